// Grouper_By_NumGroup_66099546685656
// MI455X (gfx1250) — compile-verified
//
#include <hip/hip_runtime.h>
#include <stdint.h>

#define BATCH 8
#define NPTS  16384
#define NGRP  1024
#define KNN   32
#define NT    256   // point tile per kNN block iteration
#define NTILES (NPTS / NT)

typedef __attribute__((ext_vector_type(2))) float v2f;
typedef __attribute__((ext_vector_type(8))) float v8f;
typedef __attribute__((ext_vector_type(4))) unsigned int v4u;
typedef __attribute__((ext_vector_type(8))) int v8i;
typedef __attribute__((ext_vector_type(4))) int v4i;

__device__ __forceinline__ unsigned laneid() { return threadIdx.x & 31u; }

__device__ __forceinline__ unsigned bperm32(unsigned v, int byteIdx) {
  return (unsigned)__builtin_amdgcn_ds_bpermute(byteIdx, (int)v);
}
__device__ __forceinline__ uint64_t bcast64(uint64_t v, int srcLane) {
  int bi = srcLane << 2;
  unsigned lo = bperm32((unsigned)v, bi);
  unsigned hi = bperm32((unsigned)(v >> 32), bi);
  return ((uint64_t)hi << 32) | lo;
}
__device__ __forceinline__ uint64_t shfl_xor64(uint64_t v, int off) {
  int bi = (int)((laneid() ^ (unsigned)off) << 2);
  unsigned lo = bperm32((unsigned)v, bi);
  unsigned hi = bperm32((unsigned)(v >> 32), bi);
  return ((uint64_t)hi << 32) | lo;
}
__device__ __forceinline__ unsigned ballot32(bool p) {
  return (unsigned)__builtin_amdgcn_ballot_w32(p);
}
// monotone map f32 -> u32 (total order, handles tiny negatives from rounding)
__device__ __forceinline__ unsigned fkey(float f) {
  unsigned b = __float_as_uint(f);
  return b ^ ((b & 0x80000000u) ? 0xFFFFFFFFu : 0x80000000u);
}

#if defined(__has_builtin)
#if __has_builtin(__builtin_amdgcn_tensor_load_to_lds)
#define HAVE_TDM 1
#endif
#endif

#ifdef HAVE_TDM
// 1-D TDM descriptor: copy `nfloats` consecutive f32 from global to LDS.
__device__ __forceinline__ void tdm_load_row(const float* gsrc, float* ldst,
                                             int nfloats) {
  unsigned lds_addr = (unsigned)(uintptr_t)ldst;         // low 32 bits = LDS off
  unsigned long long ga = (unsigned long long)(uintptr_t)gsrc;
  v4u g0;
  g0[0] = 1u;                                            // count=1 (user mode)
  g0[1] = lds_addr;                                      // lds_addr [63:32]
  g0[2] = (unsigned)ga;                                  // global_addr[31:0]
  g0[3] = (unsigned)((ga >> 32) & 0x01FFFFFFu) | (2u << 30);  // [56:32]|type=2
  v8i g1;
  g1[0] = 0x00020000;                    // wg_mask=0, data_size=2 (4 bytes)
  g1[1] = (int)((unsigned)nfloats << 16);// tensor_dim0[15:0] @ bits 63:48
  g1[2] = (int)(1u << 16);               // tensor_dim1 = 1
  g1[3] = (int)((unsigned)nfloats << 16);// tile_dim0 @ bits 127:112
  g1[4] = 1;                             // tile_dim1 = 1, tile_dim2 = 0
  g1[5] = nfloats;                       // tensor_dim0_stride[31:0]
  g1[6] = 0;
  g1[7] = 0;
  v4i z4 = {0, 0, 0, 0};
#if __clang_major__ >= 23
  v8i z8 = {0, 0, 0, 0, 0, 0, 0, 0};
  __builtin_amdgcn_tensor_load_to_lds(g0, g1, z4, z4, z8, 0);
#else
  __builtin_amdgcn_tensor_load_to_lds(g0, g1, z4, z4, 0);
#endif
}
#endif

// ---------------- Kernel 1: farthest point sampling (1 block / batch) --------
__global__ __launch_bounds__(1024)
void fps_kernel(const float* __restrict__ xyz, float* __restrict__ out_cen,
                float* __restrict__ wsx, float* __restrict__ wsy,
                float* __restrict__ wsz) {
  __shared__ float s_lp[3];
  __shared__ unsigned long long s_key;

  const int b = blockIdx.x;
  const int t = threadIdx.x;
  const float* base = xyz + (size_t)b * NPTS * 3;

  float px[16], py[16], pz[16], md[16];
#pragma unroll
  for (int j = 0; j < 16; ++j) {
    int p = t + j * 1024;                 // strided ownership
    px[j] = base[p * 3 + 0];
    py[j] = base[p * 3 + 1];
    pz[j] = base[p * 3 + 2];
    md[j] = 1e10f;
  }
  if (t == 0) {                           // center 0 = point 0 (reference start)
    float cx = px[0], cy = py[0], cz = pz[0];
    int gi = b * NGRP;
    out_cen[gi * 3 + 0] = cx; out_cen[gi * 3 + 1] = cy; out_cen[gi * 3 + 2] = cz;
    wsx[gi] = cx; wsy[gi] = cy; wsz[gi] = cz;
    s_lp[0] = cx; s_lp[1] = cy; s_lp[2] = cz;
    s_key = 0ull;
  }
  __syncthreads();

  for (int g = 1; g < NGRP; ++g) {
    float lx = s_lp[0], ly = s_lp[1], lz = s_lp[2];
    unsigned long long kmax = 0ull;
#pragma unroll
    for (int j = 0; j < 16; ++j) {
      float dx = px[j] - lx, dy = py[j] - ly, dz = pz[j] - lz;
      float d = fmaf(dz, dz, fmaf(dy, dy, dx * dx));
      float m = fminf(md[j], d);
      md[j] = m;
      unsigned p = (unsigned)(t + j * 1024);
      // value-major key, ~index as tie-break -> argmax picks smallest index
      unsigned long long key =
          ((unsigned long long)__float_as_uint(m) << 32) | (0xFFFFFFFFu - p);
      kmax = key > kmax ? key : kmax;
    }
    atomicMax(&s_key, kmax);              // ds_max_u64 block argmax
    __syncthreads();
    unsigned long long kw = s_key;
    __syncthreads();
    unsigned idx = 0xFFFFFFFFu - (unsigned)kw;
    if ((idx & 1023u) == (unsigned)t) {   // owner publishes the new center
      int j = (int)(idx >> 10);
      float cx = px[j], cy = py[j], cz = pz[j];
      s_lp[0] = cx; s_lp[1] = cy; s_lp[2] = cz;
      int gi = b * NGRP + g;
      out_cen[gi * 3 + 0] = cx; out_cen[gi * 3 + 1] = cy; out_cen[gi * 3 + 2] = cz;
      wsx[gi] = cx; wsy[gi] = cy; wsz[gi] = cz;
    }
    if (t == 0) s_key = 0ull;
    __syncthreads();
  }
}

// ---------------- Kernel 2: kNN grouping via WMMA f32 16x16x4 + TDM ----------
// Block = 512 threads (16 waves) handles 16 centers of one batch.
// Double-buffered point tiles streamed to LDS by the Tensor Data Mover.
__global__ __launch_bounds__(512)
void knn_kernel(const float* __restrict__ xyz, const float* __restrict__ wsx,
                const float* __restrict__ wsy, const float* __restrict__ wsz,
                float* __restrict__ out_nei) {
  __shared__ float spt[2][NT * 3];        // interleaved [n][xyz] tiles
  __shared__ float scx[16], scy[16], scz[16], sc2[16];
  __shared__ float dbuf[16 * NT];

  const int bid = blockIdx.x;
  const int b = bid >> 6;                 // 64 center-tiles per batch
  const int g0 = (bid & 63) * 16;
  const int t = threadIdx.x;
  const int w = t >> 5;                   // wave id 0..15, owns center g0+w
  const unsigned lane = laneid();
  const bool lo = lane < 16u;
  const float* base = xyz + (size_t)b * NPTS * 3;

  if (t < 16) {
    int gi = b * NGRP + g0 + t;
    float cx = wsx[gi], cy = wsy[gi], cz = wsz[gi];
    scx[t] = cx; scy[t] = cy; scz[t] = cz;
    sc2[t] = fmaf(cz, cz, fmaf(cy, cy, cx * cx));
  }
  __syncthreads();

  // A = centers 16x4 (K padded with 0). ISA layout: lanes 0-15 hold K=0,1;
  // lanes 16-31 hold K=2,3 for row m = lane&15. Branch-free packing.
  v2f A;
  {
    unsigned m = lane & 15u;
    float ax = scx[m], ay = scy[m], az = scz[m];
    A.x = lo ? ax : az;
    A.y = lo ? ay : 0.0f;
  }
  float c2s[8];                           // c^2 selected per D-register row
#pragma unroll
  for (int j = 0; j < 8; ++j) c2s[j] = lo ? sc2[j] : sc2[j + 8];

  uint64_t best = ~0ull;                  // lane-slot of distributed top-32
  uint64_t tau = ~0ull;                   // uniform: worst of current best-32
  int wl = 0;                             // lane holding tau

#ifdef HAVE_TDM
  if (w == 0) tdm_load_row(base, &spt[0][0], NT * 3);   // prologue: tile 0
#endif

  for (int tile = 0; tile < NTILES; ++tile) {
    const int cur = tile & 1;
#ifdef HAVE_TDM
    if (w == 0) {                         // issue next tile, wait current
      if (tile + 1 < NTILES) {
        tdm_load_row(base + (size_t)(tile + 1) * NT * 3, &spt[cur ^ 1][0],
                     NT * 3);
        __builtin_amdgcn_s_wait_tensorcnt(1);   // oldest (tile) complete
      } else {
        __builtin_amdgcn_s_wait_tensorcnt(0);
      }
    }
#else
    if (t < NT) {
      size_t p = (size_t)(tile * NT + t) * 3;
      spt[cur][t * 3 + 0] = base[p];
      spt[cur][t * 3 + 1] = base[p + 1];
      spt[cur][t * 3 + 2] = base[p + 2];
    }
#endif
    __syncthreads();                      // tile `cur` visible to all waves

    // wave w computes dot tile for points n in [w*16, w*16+16)
    {
      int n = w * 16 + (int)(lane & 15u);
      const float* pp = &spt[cur][n * 3];
      float xx = pp[0], yy = pp[1], zz = pp[2];
      v2f Bm;                             // B 4x16, K padded with 0
      Bm.x = lo ? xx : zz;
      Bm.y = lo ? yy : 0.0f;
      float x2l = fmaf(zz, zz, fmaf(yy, yy, xx * xx));
      v8f C = {};
      v8f D = __builtin_amdgcn_wmma_f32_16x16x4_f32(
          false, A, false, Bm, (short)0, C, false, false);
      int mofs = lo ? 0 : 8;
#pragma unroll
      for (int j = 0; j < 8; ++j) {
        float dist = fmaf(D[j], -2.0f, c2s[j] + x2l);   // |c|^2+|p|^2-2c.p
        dbuf[(j + mofs) * NT + n] = dist;
      }
    }
    __syncthreads();

    // selection: wave w streams its distance row; threshold + serialized insert
    int ibase = tile * NT;
#pragma unroll
    for (int k = 0; k < NT / 32; ++k) {
      int n = (int)lane + 32 * k;
      float d = dbuf[w * NT + n];
      uint64_t ck = ((uint64_t)fkey(d) << 32) | (unsigned)(ibase + n);
      unsigned m = ballot32(ck < tau);
      while (m) {
        int s = (int)__builtin_ctz(m); m &= m - 1u;
        uint64_t c = bcast64(ck, s);
        if (c < tau) {                    // uniform branch
          if ((int)lane == wl) best = c;  // replace current worst slot
          uint64_t mx = best;
#pragma unroll
          for (int off = 16; off; off >>= 1) {
            uint64_t o = shfl_xor64(mx, off);
            mx = (o > mx) ? o : mx;
          }
          tau = mx;
          wl = (int)__builtin_ctz(ballot32(best == mx));
        }
      }
    }
    // no trailing barrier needed: next iteration's post-issue barrier
    // separates these dbuf reads from the next compute-phase writes, and the
    // TDM overwrite target (buf[(tile+1)^1]) was last read before this
    // iteration's second barrier.
  }

  // sort best-32 ascending across lanes (matches top_k order incl. index ties)
  uint64_t key = best;
  uint64_t outk = 0;
  for (int k = 0; k < KNN; ++k) {
    uint64_t mn = key;
#pragma unroll
    for (int off = 16; off; off >>= 1) {
      uint64_t o = shfl_xor64(mn, off);
      mn = (o < mn) ? o : mn;
    }
    int src = (int)__builtin_ctz(ballot32(key == mn));
    if ((int)lane == src) key = ~0ull;
    if ((int)lane == k) outk = mn;
  }
  int idx = (int)(outk & 0xFFFFFFFFu);
  float ccx = scx[w], ccy = scy[w], ccz = scz[w];
  size_t p = (size_t)idx * 3;
  float x = base[p], y = base[p + 1], z = base[p + 2];
  size_t o = ((size_t)((b * NGRP + g0 + w) * KNN + (int)lane)) * 3;
  out_nei[o + 0] = x - ccx;
  out_nei[o + 1] = y - ccy;
  out_nei[o + 2] = z - ccz;
}

// ---------------------------------------------------------------------------
extern "C" void kernel_launch(void* const* d_in, const int* in_sizes, int n_in,
                              void* d_out, int out_size, void* d_ws,
                              size_t ws_size, hipStream_t stream) {
  (void)in_sizes; (void)n_in; (void)out_size; (void)ws_size;
  const float* xyz = (const float*)d_in[0];
  float* out = (float*)d_out;
  float* out_nei = out;                                   // [B,G,K,3]
  float* out_cen = out + (size_t)BATCH * NGRP * KNN * 3;  // [B,G,3]
  float* wsx = (float*)d_ws;
  float* wsy = wsx + BATCH * NGRP;
  float* wsz = wsy + BATCH * NGRP;

  fps_kernel<<<dim3(BATCH), dim3(1024), 0, stream>>>(xyz, out_cen, wsx, wsy, wsz);
  knn_kernel<<<dim3(BATCH * (NGRP / 16)), dim3(512), 0, stream>>>(
      xyz, wsx, wsy, wsz, out_nei);
}